// BiGRUEncoder_51814485459462
// MI455X (gfx1250) — compile-verified
//
#include <hip/hip_runtime.h>
#include <cmath>

typedef __attribute__((ext_vector_type(2))) float v2f;
typedef __attribute__((ext_vector_type(8))) float v8f;
typedef int v2i_t __attribute__((vector_size(8)));
typedef int v4i_t __attribute__((vector_size(16)));

namespace {
constexpr int BB = 32;     // batch
constexpr int TT = 1024;   // time
constexpr int NN = 512;    // hidden
constexpr int G3 = 1536;   // 3*N
}

// ---- CDNA5 async global->LDS copy (ASYNCcnt) with safe fallback -----------
#if __has_builtin(__builtin_amdgcn_global_load_async_to_lds_b128) && \
    __has_builtin(__builtin_amdgcn_global_load_async_to_lds_b64)  && \
    __has_builtin(__builtin_amdgcn_s_wait_asynccnt)
#define ASYNC_CP128(dst, src)                                         \
    __builtin_amdgcn_global_load_async_to_lds_b128(                   \
        (__attribute__((address_space(1))) v4i_t*)(src),              \
        (__attribute__((address_space(3))) v4i_t*)(dst), 0, 0)
#define ASYNC_CP64(dst, src)                                          \
    __builtin_amdgcn_global_load_async_to_lds_b64(                    \
        (__attribute__((address_space(1))) v2i_t*)(src),              \
        (__attribute__((address_space(3))) v2i_t*)(dst), 0, 0)
#define ASYNC_WAIT(N) __builtin_amdgcn_s_wait_asynccnt(N)
#else
#define ASYNC_CP128(dst, src) (*(float4*)(dst) = *(const float4*)(src))
#define ASYNC_CP64(dst, src)  (*(v2f*)(dst)   = *(const v2f*)(src))
#define ASYNC_WAIT(N)
#endif

// ---------------------------------------------------------------------------
// Phase 1: gi[dir][(t*B+b)][3N] = x_dir[t,b,:] @ Wih_dir^T
//   x_F[t,b,:] = x[b,t,:],  x_B[t,b,:] = x[b,T-1-t,:]
// grid (2048 m-tiles of 16 rows, 12 n-tiles of 128 cols, 2 dirs), 256 thr.
// K=512 in 16 chunks of 32, double-buffered async staging into LDS.
// Every thread issues exactly 5 async ops per chunk (1x b64 A + 4x b128 B),
// so s_wait_asynccnt(5) == "previous chunk fully landed".
// ---------------------------------------------------------------------------
__global__ __launch_bounds__(256)
void gi_gemm_kernel(const float* __restrict__ x,
                    const float* __restrict__ WihF,
                    const float* __restrict__ WihB,
                    float* __restrict__ gi)
{
    const int tid  = threadIdx.x;
    const int lane = tid & 31;
    const int wave = tid >> 5;
    const int l    = lane & 15;
    const int half = lane >> 4;

    const int mtile = blockIdx.x;     // 16 rows of (t,b)
    const int ntile = blockIdx.y;     // 128 output cols
    const int dir   = blockIdx.z;

    const float* __restrict__ Wih = dir ? WihB : WihF;
    float* __restrict__ giD = gi + (size_t)dir * TT * BB * G3;

    // stride 36 floats: conflict-free frag loads, 144B rows (16B aligned)
    __shared__ float A_s[2][16][36];
    __shared__ float B_s[2][128][36];

    const int colbase = ntile * 128;

    // A-source row for this thread (uniform staging, 5 async ops/chunk/thread)
    const int a_row = tid >> 4;          // 0..15
    const int a_kp  = tid & 15;          // k-pair within chunk (2 floats)
    size_t a_base;
    {
        int g = mtile * 16 + a_row;
        int t = g >> 5, b = g & 31;
        int t_eff = dir ? (TT - 1 - t) : t;
        a_base = ((size_t)b * TT + t_eff) * NN;
    }

    auto stage = [&](int buf, int k0) {
        // A: 16 rows x 32 k  (one b64 per thread)
        ASYNC_CP64(&A_s[buf][a_row][a_kp * 2], &x[a_base + k0 + a_kp * 2]);
        // B: 128 rows x 32 k (4x b128 per thread)
        #pragma unroll
        for (int s = 0; s < 4; ++s) {
            int idx = tid + s * 256;
            int row = idx >> 3;          // 8 float4 per row
            int kq  = idx & 7;
            ASYNC_CP128(&B_s[buf][row][kq * 4],
                        &Wih[(size_t)(colbase + row) * NN + k0 + kq * 4]);
        }
    };

    v8f acc = {};
    stage(0, 0);
    for (int c = 0; c < 16; ++c) {
        if (c < 15) stage((c + 1) & 1, (c + 1) * 32);
        if (c < 15) { ASYNC_WAIT(5); } else { ASYNC_WAIT(0); }
        __syncthreads();
        const int buf = c & 1;
        #pragma unroll
        for (int kk = 0; kk < 32; kk += 4) {
            v2f a = *(const v2f*)&A_s[buf][l][kk + 2 * half];
            v2f b = *(const v2f*)&B_s[buf][wave * 16 + l][kk + 2 * half];
            acc = __builtin_amdgcn_wmma_f32_16x16x4_f32(
                false, a, false, b, (short)0, acc, false, false);
        }
        __syncthreads();
    }

    // C/D layout: VGPR v -> row (v + 8*half), col = lane&15
    #pragma unroll
    for (int v = 0; v < 8; ++v) {
        int row = mtile * 16 + v + 8 * half;
        int col = colbase + wave * 16 + l;
        giD[(size_t)row * G3 + col] = acc[v];
    }
}

// ---------------------------------------------------------------------------
// Phase 2: one timestep. 64 WGs = 32 hidden-unit tiles x 2 directions.
// 8 waves stage (uniform 5 async ops / chunk / thread); waves 0..5 compute
// (m-tile of batch x gate r/z/n). Whh stays L2-resident across all steps.
// ---------------------------------------------------------------------------
__global__ __launch_bounds__(256)
void gru_step_kernel(const float* __restrict__ x,
                     const float* __restrict__ WhhF,
                     const float* __restrict__ WhhB,
                     const float* __restrict__ bihF,
                     const float* __restrict__ bhhF,
                     const float* __restrict__ bihB,
                     const float* __restrict__ bhhB,
                     const float* __restrict__ gi,
                     float* __restrict__ hbuf,
                     float* __restrict__ out,
                     int t)
{
    const int tid  = threadIdx.x;
    const int lane = tid & 31;
    const int wave = tid >> 5;     // 0..7 (0..5 compute)
    const int l    = lane & 15;
    const int half = lane >> 4;
    const int mt   = wave & 1;     // batch half (rows 0..15 / 16..31)
    const int gate = wave >> 1;    // 0=r 1=z 2=n  (valid for wave<6)

    const int jt  = blockIdx.x & 31;   // hidden tile
    const int dir = blockIdx.x >> 5;
    const int j0  = jt * 16;

    const float* __restrict__ Whh = dir ? WhhB : WhhF;
    const float* __restrict__ bih = dir ? bihB : bihF;
    const float* __restrict__ bhh = dir ? bhhB : bhhF;
    const float* __restrict__ giD =
        gi + (size_t)dir * TT * BB * G3 + (size_t)t * BB * G3;

    float* hD = hbuf + (size_t)dir * 2 * BB * NN;
    const float* __restrict__ h_prev = hD + (size_t)(t & 1) * BB * NN;
    float* __restrict__ h_next       = hD + (size_t)((t + 1) & 1) * BB * NN;

    __shared__ float h_s[2][32][68];    // 68 % 64 == 4 -> conflict-free frags
    __shared__ float W_s[2][48][68];    // 3 gates x 16 cols
    __shared__ float exch[3][32][17];   // gh cross-wave exchange

    auto stage = [&](int buf, int kc) {
        // h chunk: 32 x 64 = 512 float4 -> 2x b128 per thread
        #pragma unroll
        for (int s = 0; s < 2; ++s) {
            int idx = tid + s * 256;
            int row = idx >> 4, kq = idx & 15;
            ASYNC_CP128(&h_s[buf][row][kq * 4],
                        &h_prev[(size_t)row * NN + kc + kq * 4]);
        }
        // Whh chunk: 48 x 64 = 768 float4 -> 3x b128 per thread
        #pragma unroll
        for (int s = 0; s < 3; ++s) {
            int idx = tid + s * 256;
            int c = idx >> 4, kq = idx & 15;
            int grow = (c >> 4) * NN + j0 + (c & 15);
            ASYNC_CP128(&W_s[buf][c][kq * 4],
                        &Whh[(size_t)grow * NN + kc + kq * 4]);
        }
    };

    v8f acc = {};
    stage(0, 0);
    for (int c = 0; c < 8; ++c) {
        if (c < 7) stage((c + 1) & 1, (c + 1) * 64);
        if (c < 7) { ASYNC_WAIT(5); } else { ASYNC_WAIT(0); }
        __syncthreads();
        if (wave < 6) {
            const int buf = c & 1;
            #pragma unroll
            for (int kk = 0; kk < 64; kk += 4) {
                v2f a = *(const v2f*)&h_s[buf][mt * 16 + l][kk + 2 * half];
                v2f b = *(const v2f*)&W_s[buf][gate * 16 + l][kk + 2 * half];
                acc = __builtin_amdgcn_wmma_f32_16x16x4_f32(
                    false, a, false, b, (short)0, acc, false, false);
            }
        }
        __syncthreads();
    }

    // gh -> LDS so every thread can see all 3 gates of its (b, j)
    if (wave < 6) {
        #pragma unroll
        for (int v = 0; v < 8; ++v) {
            int bb = mt * 16 + v + 8 * half;
            exch[gate][bb][l] = acc[v];
        }
    }
    __syncthreads();

    const int t_x = dir ? (TT - 1 - t) : t;
    for (int idx = tid; idx < 512; idx += 256) {
        int b  = idx >> 4;
        int jj = idx & 15;
        int j  = j0 + jj;
        float gh_r = exch[0][b][jj] + bhh[j];
        float gh_z = exch[1][b][jj] + bhh[NN + j];
        float gh_n = exch[2][b][jj] + bhh[2 * NN + j];
        const float* girow = &giD[(size_t)b * G3];
        float gi_r = girow[j]          + bih[j];
        float gi_z = girow[NN + j]     + bih[NN + j];
        float gi_n = girow[2 * NN + j] + bih[2 * NN + j];
        float r = 1.0f / (1.0f + __expf(-(gi_r + gh_r)));
        float z = 1.0f / (1.0f + __expf(-(gi_z + gh_z)));
        float n = tanhf(gi_n + r * gh_n);
        float hp = h_prev[(size_t)b * NN + j];
        float hn = (1.0f - z) * n + z * hp;
        h_next[(size_t)b * NN + j] = hn;
        float xv = x[((size_t)b * TT + t_x) * NN + j];
        out[((size_t)b * TT + t) * (2 * NN) + dir * NN + j] = hn + xv;
    }
}

// ---------------------------------------------------------------------------
extern "C" void kernel_launch(void* const* d_in, const int* in_sizes, int n_in,
                              void* d_out, int out_size, void* d_ws, size_t ws_size,
                              hipStream_t stream)
{
    const float* x    = (const float*)d_in[0];
    const float* WihF = (const float*)d_in[1];
    const float* WhhF = (const float*)d_in[2];
    const float* bihF = (const float*)d_in[3];
    const float* bhhF = (const float*)d_in[4];
    const float* WihB = (const float*)d_in[5];
    const float* WhhB = (const float*)d_in[6];
    const float* bihB = (const float*)d_in[7];
    const float* bhhB = (const float*)d_in[8];
    float* out = (float*)d_out;

    // workspace layout: gi (2 dirs * T*B*3N f32 ~= 403 MB) + h ping-pong
    float* gi   = (float*)d_ws;
    float* hbuf = gi + (size_t)2 * TT * BB * G3;

    // zero h(0) for both directions (both slots; keeps graph replay deterministic)
    (void)hipMemsetAsync(hbuf, 0, (size_t)2 * 2 * BB * NN * sizeof(float), stream);

    // Phase 1: all input-to-hidden GEMMs in parallel
    dim3 g1(2048, 12, 2);
    gi_gemm_kernel<<<g1, 256, 0, stream>>>(x, WihF, WihB, gi);

    // Phase 2: sequential recurrence, one small kernel per timestep
    for (int t = 0; t < TT; ++t) {
        gru_step_kernel<<<dim3(64), 256, 0, stream>>>(
            x, WhhF, WhhB, bihF, bhhF, bihB, bhhB, gi, hbuf, out, t);
    }
}